// SmallNet_21612275433880
// MI455X (gfx1250) — compile-verified
//
#include <hip/hip_runtime.h>

// ---------------- problem constants (match reference) ----------------
#define N_POINTS   100000
#define N_EVENTS   2000000
#define N_PAIRS    10
#define MC_SAMPLES 100000
#define EPS        1e-6f

// ---------------- launch shape ----------------
#define NT     256          // 8 waves of 32 on gfx1250
#define NB_EV  2048         // ~4 events/thread
#define NB_MC  1024         // ~4 samples/thread

typedef __attribute__((ext_vector_type(2))) float v2f;
typedef __attribute__((ext_vector_type(8))) float v8f;

// Full-precision 32-lane sum using V_WMMA_F32_16X16X4_F32.
// A layout (32-bit A, 16x4): lanes 0-15 hold A[m][0]/A[m][1] in vgpr0/1,
// lanes 16-31 hold A[m][2]/A[m][3]. With a=(x,0) and B=all-ones:
//   D[m][n] = x_m + x_{m+16}  (independent of B layout).
// D layout: vgpr j -> row j (lanes 0-15) / row j+8 (lanes 16-31), so
// sum(c[0..7]) on a low lane covers rows 0..7, on a high lane rows 8..15;
// one shfl_xor(16) completes the exact tree sum on every lane.
__device__ __forceinline__ float wave_sum32_wmma(float x) {
    v2f a; a[0] = x;    a[1] = 0.0f;
    v2f b; b[0] = 1.0f; b[1] = 1.0f;
    v8f c = {0.f, 0.f, 0.f, 0.f, 0.f, 0.f, 0.f, 0.f};
    c = __builtin_amdgcn_wmma_f32_16x16x4_f32(
            /*neg_a=*/false, a, /*neg_b=*/false, b,
            /*c_mod=*/(short)0, c, /*reuse_a=*/false, /*reuse_b=*/false);
    float t = ((c[0] + c[1]) + (c[2] + c[3])) + ((c[4] + c[5]) + (c[6] + c[7]));
    t += __shfl_xor(t, 16, 32);
    return t;   // valid in all 32 lanes
}

// Block (256-thread) sum: WMMA per wave, 8-float LDS hop, serial-8 on t0.
__device__ __forceinline__ float block_sum(float x, float* smem8) {
    float w = wave_sum32_wmma(x);
    const int wid  = threadIdx.x >> 5;
    const int lane = threadIdx.x & 31;
    if (lane == 0) smem8[wid] = w;
    __syncthreads();
    float r = 0.f;
    if (threadIdx.x == 0) {
        #pragma unroll
        for (int i = 0; i < NT / 32; ++i) r += smem8[i];
    }
    return r;   // valid on thread 0 only
}

// ---------------- event term:  sum over events of (beta - ||dz + eps||) ----
__global__ __launch_bounds__(NT) void event_kernel(
        const int*    __restrict__ u_idx,
        const int*    __restrict__ v_idx,
        const float*  __restrict__ et,
        const float2* __restrict__ z0,
        const float2* __restrict__ v0,
        const float2* __restrict__ a0,
        const float*  __restrict__ beta,
        float*        __restrict__ partial) {
    __shared__ float smem8[NT / 32];
    const float b = beta[0];
    float acc = 0.f;
    for (int i = blockIdx.x * NT + threadIdx.x; i < N_EVENTS; i += NB_EV * NT) {
        const int u = u_idx[i];
        const int v = v_idx[i];
        const float t   = et[i];
        const float ht2 = 0.5f * t * t;
        const float2 zu = z0[u], zv = z0[v];
        const float2 vu = v0[u], vv = v0[v];
        const float2 au = a0[u], av = a0[v];
        const float dx = (zu.x - zv.x) + (vu.x - vv.x) * t + (au.x - av.x) * ht2 + EPS;
        const float dy = (zu.y - zv.y) + (vu.y - vv.y) * t + (au.y - av.y) * ht2 + EPS;
        acc += b - sqrtf(dx * dx + dy * dy);
    }
    const float r = block_sum(acc, smem8);
    if (threadIdx.x == 0) partial[blockIdx.x] = r;
}

// ---------------- MC term: sum over 10*100000 of exp(beta - ||dz + eps||) ---
__global__ __launch_bounds__(NT) void mc_kernel(
        const int*    __restrict__ su,
        const int*    __restrict__ sv,
        const float*  __restrict__ st,       // (N_PAIRS, MC_SAMPLES) row-major
        const float2* __restrict__ z0,
        const float2* __restrict__ v0,
        const float2* __restrict__ a0,
        const float*  __restrict__ beta,
        float*        __restrict__ partial) {
    __shared__ float pd[N_PAIRS][6];         // dz0x,dz0y,dvx,dvy,0.5*dax,0.5*day
    __shared__ float smem8[NT / 32];
    if (threadIdx.x < N_PAIRS) {
        const int u = su[threadIdx.x];
        const int v = sv[threadIdx.x];
        const float2 zu = z0[u], zv = z0[v];
        const float2 vu = v0[u], vv = v0[v];
        const float2 au = a0[u], av = a0[v];
        pd[threadIdx.x][0] = zu.x - zv.x;
        pd[threadIdx.x][1] = zu.y - zv.y;
        pd[threadIdx.x][2] = vu.x - vv.x;
        pd[threadIdx.x][3] = vu.y - vv.y;
        pd[threadIdx.x][4] = 0.5f * (au.x - av.x);
        pd[threadIdx.x][5] = 0.5f * (au.y - av.y);
    }
    __syncthreads();
    const float b = beta[0];
    float acc = 0.f;
    const int total = N_PAIRS * MC_SAMPLES;
    for (int i = blockIdx.x * NT + threadIdx.x; i < total; i += NB_MC * NT) {
        const int   p  = i / MC_SAMPLES;     // constant divisor -> mul/shift
        const float t  = st[i];
        const float t2 = t * t;
        const float dx = pd[p][0] + pd[p][2] * t + pd[p][4] * t2 + EPS;
        const float dy = pd[p][1] + pd[p][3] * t + pd[p][5] * t2 + EPS;
        acc += expf(b - sqrtf(dx * dx + dy * dy));
    }
    const float r = block_sum(acc, smem8);
    if (threadIdx.x == 0) partial[blockIdx.x] = r;
}

// ---------------- finalize: single wave, exact tree over block partials ----
__global__ __launch_bounds__(32) void finalize_kernel(
        const float* __restrict__ evp, int n_ev,
        const float* __restrict__ mcp, int n_mc,
        const float* __restrict__ beta,   // unused numerically here; b folded upstream
        const float* __restrict__ t0,
        const float* __restrict__ tn,
        float*       __restrict__ out) {
    (void)beta;
    const int lane = threadIdx.x;
    float ae = 0.f, am = 0.f;
    for (int i = lane; i < n_ev; i += 32) ae += evp[i];
    for (int i = lane; i < n_mc; i += 32) am += mcp[i];
    const float E  = wave_sum32_wmma(ae);           // event_intensity
    const float SL = wave_sum32_wmma(am);           // sum of lambda over all MC samples
    if (lane == 0) {
        const float interval = tn[0] - t0[0];
        const float N = interval * (SL / (float)MC_SAMPLES);  // NON_EVENT_WEIGHT == 1
        out[0] = E - N;     // log_likelihood
        out[1] = E / N;     // ratio
    }
}

extern "C" void kernel_launch(void* const* d_in, const int* in_sizes, int n_in,
                              void* d_out, int out_size, void* d_ws, size_t ws_size,
                              hipStream_t stream) {
    (void)in_sizes; (void)n_in; (void)out_size; (void)ws_size;
    const int*    u_idx = (const int*)   d_in[0];
    const int*    v_idx = (const int*)   d_in[1];
    const float*  et    = (const float*) d_in[2];
    const int*    su    = (const int*)   d_in[3];
    const int*    sv    = (const int*)   d_in[4];
    const float*  st    = (const float*) d_in[5];
    const float*  t0    = (const float*) d_in[6];
    const float*  tn    = (const float*) d_in[7];
    const float*  beta  = (const float*) d_in[8];
    const float2* z0    = (const float2*)d_in[9];
    const float2* v0    = (const float2*)d_in[10];
    const float2* a0    = (const float2*)d_in[11];

    float* ws  = (float*)d_ws;
    float* evp = ws;            // NB_EV partials
    float* mcp = ws + NB_EV;    // NB_MC partials

    event_kernel<<<NB_EV, NT, 0, stream>>>(u_idx, v_idx, et, z0, v0, a0, beta, evp);
    mc_kernel   <<<NB_MC, NT, 0, stream>>>(su, sv, st, z0, v0, a0, beta, mcp);
    finalize_kernel<<<1, 32, 0, stream>>>(evp, NB_EV, mcp, NB_MC, beta, t0, tn,
                                          (float*)d_out);
}